// MemoryLayer_8478265442370
// MI455X (gfx1250) — compile-verified
//
#include <hip/hip_runtime.h>
#include <math.h>

// Problem constants (from the reference)
#define B_   32
#define T_   64
#define I_   256
#define H_   512
#define N2H  1024                      // 2*H
#define ALPHA_ 0.9512294245007140f     // exp(-1/20)
#define ETA_   0.1f

typedef float v2f __attribute__((ext_vector_type(2)));
typedef float v8f __attribute__((ext_vector_type(8)));

// ---------------------------------------------------------------------------
// CDNA5 f32 WMMA: D(16x16,f32) = A(16x4,f32) x B(4x16,f32) + C
// Lane layout (ISA 7.12.2):
//   A: lane (l%16)=M row, VGPR v holds K = 2*(l/16)+v
//   B: lane (l%16)=N col, VGPR v holds K = 2*(l/16)+v
//   D: VGPR r holds M = r + 8*(l/16), N = l%16
// ---------------------------------------------------------------------------
__device__ __forceinline__ v8f wmma_f32(v2f a, v2f b, v8f c) {
  return __builtin_amdgcn_wmma_f32_16x16x4_f32(
      /*neg_a=*/false, a, /*neg_b=*/false, b,
      /*c_mod=*/(short)0, c, /*reuse_a=*/false, /*reuse_b=*/false);
}

// ===========================================================================
// Kernel 1: pre = x @ W^T      [2048 x 1024], K = 256, f32 WMMA
// Each wave: one 16(M) x 64(N) strip -> A loaded once per k-step, 4 WMMAs.
// ===========================================================================
__global__ __launch_bounds__(256) void k_pregemm(const float* __restrict__ x,
                                                 const float* __restrict__ W,
                                                 float* __restrict__ pre) {
  const int lane = threadIdx.x & 31;
  const int gw   = blockIdx.x * 8 + (threadIdx.x >> 5);   // 2048 waves total
  const int m0   = (gw >> 4) << 4;                        // M tile (x row = b*T+t)
  const int n0   = (gw & 15) << 6;                        // N strip of 64
  const int row  = lane & 15;
  const int koff = (lane >> 4) << 1;                      // 0 or 2

  v8f acc[4] = {};
  const float* xrow = x + (size_t)(m0 + row) * I_;
  for (int k0 = 0; k0 < I_; k0 += 4) {
    v2f a = *(const v2f*)(xrow + k0 + koff);
#pragma unroll
    for (int s = 0; s < 4; ++s) {
      v2f b = *(const v2f*)(W + (size_t)(n0 + s * 16 + row) * I_ + k0 + koff);
      acc[s] = wmma_f32(a, b, acc[s]);
    }
  }
  const int mbase = m0 + ((lane >> 4) << 3);
#pragma unroll
  for (int s = 0; s < 4; ++s)
#pragma unroll
    for (int r = 0; r < 8; ++r)
      pre[(size_t)(mbase + r) * N2H + n0 + s * 16 + row] = acc[s][r];
}

// ===========================================================================
// Kernel 2: sequential scan (one block of 512 threads per batch).
// Per-thread (h-index j) state kv,vv,kt,vt lives in VGPRs.
// ikv_t[j] = 0.2*eta * sum_{s<t} vt_s[j] * <kt_s, k_t>  (rank-<=64 trick).
// kt/vt histories in LDS (~260 KB of the 320 KB WGP LDS).
// ===========================================================================
#define KSTR (H_ + 8)   // padded stride: 8-lane dot groups hit distinct banks
#define SCAN_LDS_BYTES ((T_ * KSTR + T_ * H_ + H_ + T_) * sizeof(float))

__global__ __launch_bounds__(512) void k_scan(const float* __restrict__ pre,
                                              float* __restrict__ keys,
                                              float* __restrict__ vals,
                                              float* __restrict__ KT,
                                              float* __restrict__ VT) {
  extern __shared__ float smem[];
  float* kt_hist = smem;                    // [T][KSTR]
  float* vt_hist = kt_hist + T_ * KSTR;     // [T][H]
  float* k_sh    = vt_hist + T_ * H_;       // [H]
  float* c_sh    = k_sh + H_;               // [T]

  const int b  = blockIdx.x;
  const int j  = threadIdx.x;               // h index 0..511
  const int sg = j >> 3;                    // 8-lane dot group -> history row s
  const int l8 = j & 7;

  const float OMD = 1.0f - ALPHA_;
  float kv = 0.f, vv = 0.f, kt = 0.f, vt = 0.f;

  for (int t = 0; t < T_; ++t) {
    const size_t bt = (size_t)(b * T_ + t);
    const float ik = pre[bt * N2H + j];
    const float iv = pre[bt * N2H + H_ + j];

    kv = ALPHA_ * kv + ik;
    const float k = tanhf(kv);
    k_sh[j] = k;
    keys[bt * H_ + j] = k;
    __syncthreads();                        // k_sh + prev-step histories visible

    // c_s = 0.2*eta * <kt_s, k_t> for s < t (8 lanes per s, wave32 shuffles)
    if (sg < t) {
      const float* krow = kt_hist + sg * KSTR;
      float p = 0.f;
      for (int jj = l8; jj < H_; jj += 8) p += krow[jj] * k_sh[jj];
      p += __shfl_xor(p, 1);
      p += __shfl_xor(p, 2);
      p += __shfl_xor(p, 4);
      if (l8 == 0) c_sh[sg] = 0.02f * p;    // 0.2 * ETA
    }
    __syncthreads();

    float ikv = 0.f;
    for (int s = 0; s < t; ++s) ikv += vt_hist[s * H_ + j] * c_sh[s];

    vv = ALPHA_ * vv + iv + ikv;
    const float v = tanhf(vv);
    vals[bt * H_ + j] = v;

    kt = ALPHA_ * kt + OMD * k;
    vt = ALPHA_ * vt + OMD * v;
    kt_hist[t * KSTR + j] = kt;
    vt_hist[t * H_ + j]   = vt;
    KT[bt * H_ + j] = kt;
    VT[bt * H_ + j] = vt;
    __syncthreads();                        // protect k_sh/c_sh for next step
  }
}

// ===========================================================================
// Kernel 3: mem[b] = ETA * VT[b]^T @ KT[b]   (512x512, K=64) per batch.
// Each wave: 16(M) x 64(N) strip. Streams the 32 MB output with NT stores.
// ===========================================================================
__global__ __launch_bounds__(256) void k_memgemm(const float* __restrict__ KT,
                                                 const float* __restrict__ VT,
                                                 float* __restrict__ mem) {
  const int lane = threadIdx.x & 31;
  const int gw   = blockIdx.x * 8 + (threadIdx.x >> 5);   // 8192 waves total
  const int b    = gw >> 8;                               // 256 wave-strips/batch
  const int rem  = gw & 255;
  const int m0   = (rem >> 3) << 4;
  const int n0   = (rem & 7) << 6;
  const int row  = lane & 15;
  const int koff = (lane >> 4) << 1;

  const float* vtb = VT + (size_t)b * T_ * H_;
  const float* ktb = KT + (size_t)b * T_ * H_;

  v8f acc[4] = {};
  for (int k0 = 0; k0 < T_; k0 += 4) {
    v2f a;
    a.x = vtb[(size_t)(k0 + koff + 0) * H_ + m0 + row];
    a.y = vtb[(size_t)(k0 + koff + 1) * H_ + m0 + row];
#pragma unroll
    for (int s = 0; s < 4; ++s) {
      v2f bb;
      bb.x = ktb[(size_t)(k0 + koff + 0) * H_ + n0 + s * 16 + row];
      bb.y = ktb[(size_t)(k0 + koff + 1) * H_ + n0 + s * 16 + row];
      acc[s] = wmma_f32(a, bb, acc[s]);
    }
  }
  float* mb = mem + (size_t)b * H_ * H_;
  const int mbase = m0 + ((lane >> 4) << 3);
#pragma unroll
  for (int s = 0; s < 4; ++s)
#pragma unroll
    for (int r = 0; r < 8; ++r)
      __builtin_nontemporal_store(ETA_ * acc[s][r],
                                  mb + (size_t)(mbase + r) * H_ + n0 + s * 16 + row);
}

// ===========================================================================
extern "C" void kernel_launch(void* const* d_in, const int* in_sizes, int n_in,
                              void* d_out, int out_size, void* d_ws, size_t ws_size,
                              hipStream_t stream) {
  (void)in_sizes; (void)n_in; (void)out_size; (void)ws_size;
  const float* x = (const float*)d_in[0];   // [B,T,I]
  const float* W = (const float*)d_in[1];   // [2H,I]

  float* out  = (float*)d_out;
  float* mem  = out;                                   // [B,H,H]
  float* keys = out + (size_t)B_ * H_ * H_;            // [B,T,H]
  float* vals = keys + (size_t)B_ * T_ * H_;           // [B,T,H]

  float* pre = (float*)d_ws;                           // [B,T,2H]  8 MB
  float* KT  = pre + (size_t)B_ * T_ * N2H;            // [B,T,H]   4 MB
  float* VT  = KT + (size_t)B_ * T_ * H_;              // [B,T,H]   4 MB

  // 1) pre-activations: 2048x1024x256 f32 WMMA GEMM (2048 waves / 256 blocks)
  k_pregemm<<<256, 256, 0, stream>>>(x, W, pre);
  // 2) sequential scan: 1 block per batch, histories in LDS
  k_scan<<<B_, 512, SCAN_LDS_BYTES, stream>>>(pre, keys, vals, KT, VT);
  // 3) mem = eta * VT^T @ KT per batch: 32x(512x512x64) f32 WMMA (1024 blocks)
  k_memgemm<<<B_ * 32, 256, 0, stream>>>(KT, VT, mem);
}